// ODEFunc_32873679683756
// MI455X (gfx1250) — compile-verified
//
#include <hip/hip_runtime.h>

#define N_NODES 100000
#define N_EDGES 1600000
#define DIM 64

typedef __attribute__((ext_vector_type(2))) float v2f;
typedef __attribute__((ext_vector_type(8))) float v8f;

__device__ __forceinline__ v8f wmma_f32(v2f a, v2f b, v8f c) {
    // V_WMMA_F32_16X16X4_F32: full-precision f32 matrix core op (wave32).
    // 8 args: (neg_a, A, neg_b, B, c_mod, C, reuse_a, reuse_b)
    return __builtin_amdgcn_wmma_f32_16x16x4_f32(false, a, false, b, (short)0, c,
                                                 false, false);
}

// ---------------------------------------------------------------------------
// Zero a float region (harness poisons d_ws with 0xAA).
// ---------------------------------------------------------------------------
__global__ void zero_f32(float* __restrict__ p, size_t n) {
    size_t i = (size_t)blockIdx.x * blockDim.x + threadIdx.x;
    size_t stride = (size_t)gridDim.x * blockDim.x;
    for (; i < n; i += stride) p[i] = 0.0f;
}

// ---------------------------------------------------------------------------
// x = h @ W^T for both graphs (blockIdx.y selects pos/neg).
// One wave -> 16 rows x 64 cols, 16 k-chunks x 4 n-tiles = 64 WMMA ops.
// ---------------------------------------------------------------------------
__global__ void __launch_bounds__(256)
gemm_xw(const float* __restrict__ h,
        const float* __restrict__ Wpos, const float* __restrict__ Wneg,
        float* __restrict__ xpos, float* __restrict__ xneg) {
    const float* W = (blockIdx.y == 0) ? Wpos : Wneg;
    float* out = (blockIdx.y == 0) ? xpos : xneg;

    const int wave = threadIdx.x >> 5;
    const int lane = threadIdx.x & 31;
    const int tile = blockIdx.x * 8 + wave;
    if (tile * 16 >= N_NODES) return;  // uniform per wave: EXEC stays all-ones

    const int row0 = tile * 16;
    const int half = lane >> 4;   // 0: K pair {0,1}, 1: K pair {2,3} of the chunk
    const int mrow = lane & 15;

    v8f acc0 = {}, acc1 = {}, acc2 = {}, acc3 = {};
    const float* arow = h + (size_t)(row0 + mrow) * DIM;

#pragma unroll
    for (int kc = 0; kc < 16; ++kc) {
        const int k0 = kc * 4 + half * 2;
        v2f a  = *(const v2f*)(arow + k0);
        v2f b0 = *(const v2f*)(W + (size_t)(0 * 16 + mrow) * DIM + k0);
        v2f b1 = *(const v2f*)(W + (size_t)(1 * 16 + mrow) * DIM + k0);
        v2f b2 = *(const v2f*)(W + (size_t)(2 * 16 + mrow) * DIM + k0);
        v2f b3 = *(const v2f*)(W + (size_t)(3 * 16 + mrow) * DIM + k0);
        acc0 = wmma_f32(a, b0, acc0);
        acc1 = wmma_f32(a, b1, acc1);
        acc2 = wmma_f32(a, b2, acc2);
        acc3 = wmma_f32(a, b3, acc3);
    }

#pragma unroll
    for (int r = 0; r < 8; ++r) {
        const int m = row0 + half * 8 + r;  // C/D layout: vgpr r, lane half
        float* orow = out + (size_t)m * DIM + mrow;
        orow[0]  = acc0[r];
        orow[16] = acc1[r];
        orow[32] = acc2[r];
        orow[48] = acc3[r];
    }
}

// ---------------------------------------------------------------------------
// In-degree counts via f32 atomics (deg buffer pre-zeroed).
// ---------------------------------------------------------------------------
__global__ void degree_k(const int* __restrict__ ei, float* __restrict__ deg) {
    int i = blockIdx.x * blockDim.x + threadIdx.x;
    int stride = gridDim.x * blockDim.x;
    for (int e = i; e < N_EDGES; e += stride)
        atomicAdd(&deg[ei[N_EDGES + e]], 1.0f);
}

// deg -> dinv = rsqrt(deg + 1)   (self-loop contributes the +1)
__global__ void rsqrt_k(float* __restrict__ d, int n) {
    int i = blockIdx.x * blockDim.x + threadIdx.x;
    if (i < n) d[i] = rsqrtf(d[i] + 1.0f);
}

// ---------------------------------------------------------------------------
// Edge scatter: agg[dst] += x[src] * dinv[src]*dinv[dst].
// One wave per edge; each lane moves a float2 of the 64-wide row (coalesced
// 256B gather, atomics land in L2 — whole working set fits in 192MB L2).
// Edge indices are wave-uniform: force them scalar with readfirstlane so the
// index math runs on the SALU and the gather uses the saddr addressing form.
// ---------------------------------------------------------------------------
__global__ void __launch_bounds__(256)
scatter_k(const float* __restrict__ x, const float* __restrict__ dinv,
          const int* __restrict__ ei, float* __restrict__ agg) {
    const int lane = threadIdx.x & 31;
    const int wid = __builtin_amdgcn_readfirstlane(
        blockIdx.x * (blockDim.x >> 5) + (threadIdx.x >> 5));
    const int nwaves = (gridDim.x * blockDim.x) >> 5;
    for (int e = wid; e < N_EDGES; e += nwaves) {
        const int s = __builtin_amdgcn_readfirstlane(ei[e]);
        const int d = __builtin_amdgcn_readfirstlane(ei[N_EDGES + e]);
        const float norm = dinv[s] * dinv[d];
        const float2 v = *(const float2*)&x[(size_t)s * DIM + lane * 2];
        float* dst = &agg[(size_t)d * DIM + lane * 2];
        atomicAdd(dst + 0, v.x * norm);
        atomicAdd(dst + 1, v.y * norm);
    }
}

// ---------------------------------------------------------------------------
// Final fused pass:
//   y_g[m][k] = agg_g[m][k] + x_g[m][k]*dinv_g[m]^2 + b_g[k]   (g = pos,neg)
//   delta     = y_pos @ Wp^T + y_neg @ Wn^T + (bp_psi + bn_psi)
// Self-loop + GCN bias are applied while building the WMMA A-fragments,
// saving a full elementwise read-modify-write pass over both agg buffers.
// ---------------------------------------------------------------------------
__global__ void __launch_bounds__(256)
gemm_final(const float* __restrict__ aggp, const float* __restrict__ aggn,
           const float* __restrict__ xp, const float* __restrict__ xn,
           const float* __restrict__ dinvp, const float* __restrict__ dinvn,
           const float* __restrict__ bp, const float* __restrict__ bn,
           const float* __restrict__ Wp, const float* __restrict__ Wn,
           const float* __restrict__ bpsip, const float* __restrict__ bpsin,
           float* __restrict__ out) {
    const int wave = threadIdx.x >> 5;
    const int lane = threadIdx.x & 31;
    const int tile = blockIdx.x * 8 + wave;
    if (tile * 16 >= N_NODES) return;

    const int row0 = tile * 16;
    const int half = lane >> 4;
    const int mrow = lane & 15;
    const int m = row0 + mrow;

    v8f acc[4];
#pragma unroll
    for (int nt = 0; nt < 4; ++nt) {
        const float bsum = bpsip[nt * 16 + mrow] + bpsin[nt * 16 + mrow];
        acc[nt] = (v8f){bsum, bsum, bsum, bsum, bsum, bsum, bsum, bsum};
    }

    const float dp = dinvp[m];
    const float dn = dinvn[m];
    const float sp = dp * dp;   // self-loop norm, pos graph
    const float sn = dn * dn;   // self-loop norm, neg graph

    const float* ap_row = aggp + (size_t)m * DIM;
    const float* an_row = aggn + (size_t)m * DIM;
    const float* xp_row = xp + (size_t)m * DIM;
    const float* xn_row = xn + (size_t)m * DIM;

#pragma unroll
    for (int kc = 0; kc < 16; ++kc) {
        const int k0 = kc * 4 + half * 2;
        v2f a1 = *(const v2f*)(ap_row + k0);
        v2f a2 = *(const v2f*)(an_row + k0);
        v2f x1 = *(const v2f*)(xp_row + k0);
        v2f x2 = *(const v2f*)(xn_row + k0);
        v2f bb1 = *(const v2f*)(bp + k0);
        v2f bb2 = *(const v2f*)(bn + k0);
        a1 += x1 * sp + bb1;   // agg + self-loop + GCN bias (pos)
        a2 += x2 * sn + bb2;   // agg + self-loop + GCN bias (neg)
#pragma unroll
        for (int nt = 0; nt < 4; ++nt) {
            v2f b1 = *(const v2f*)(Wp + (size_t)(nt * 16 + mrow) * DIM + k0);
            v2f b2 = *(const v2f*)(Wn + (size_t)(nt * 16 + mrow) * DIM + k0);
            acc[nt] = wmma_f32(a1, b1, acc[nt]);
            acc[nt] = wmma_f32(a2, b2, acc[nt]);
        }
    }

#pragma unroll
    for (int r = 0; r < 8; ++r) {
        const int mm = row0 + half * 8 + r;
        float* orow = out + (size_t)mm * DIM + mrow;
        orow[0]  = acc[0][r];
        orow[16] = acc[1][r];
        orow[32] = acc[2][r];
        orow[48] = acc[3][r];
    }
}

extern "C" void kernel_launch(void* const* d_in, const int* in_sizes, int n_in,
                              void* d_out, int out_size, void* d_ws, size_t ws_size,
                              hipStream_t stream) {
    // setup_inputs order:
    // 0:t 1:h 2:edge_index_pos 3:edge_index_neg 4:W_pos 5:b_pos 6:W_neg 7:b_neg
    // 8:W_psi_pos 9:b_psi_pos 10:W_psi_neg 11:b_psi_neg
    const float* h         = (const float*)d_in[1];
    const int*   eip       = (const int*)d_in[2];
    const int*   ein       = (const int*)d_in[3];
    const float* W_pos     = (const float*)d_in[4];
    const float* b_pos     = (const float*)d_in[5];
    const float* W_neg     = (const float*)d_in[6];
    const float* b_neg     = (const float*)d_in[7];
    const float* W_psi_pos = (const float*)d_in[8];
    const float* b_psi_pos = (const float*)d_in[9];
    const float* W_psi_neg = (const float*)d_in[10];
    const float* b_psi_neg = (const float*)d_in[11];
    float* out = (float*)d_out;

    float* ws = (float*)d_ws;
    const size_t ND = (size_t)N_NODES * DIM;
    float* x_pos    = ws;            // [N, 64]
    float* x_neg    = ws + ND;       // [N, 64]
    float* agg_pos  = ws + 2 * ND;   // [N, 64]  (zeroed)
    float* agg_neg  = ws + 3 * ND;   // [N, 64]  (zeroed)
    float* dinv_pos = ws + 4 * ND;   // [N]      (deg then dinv, zeroed)
    float* dinv_neg = dinv_pos + N_NODES;

    // 1) zero agg_pos, agg_neg, deg buffers (contiguous region)
    const size_t zcount = 2 * ND + 2 * (size_t)N_NODES;
    zero_f32<<<2048, 256, 0, stream>>>(agg_pos, zcount);

    // 2) x = h @ W^T  (both graphs, WMMA f32)
    dim3 gx(782, 2);
    gemm_xw<<<gx, 256, 0, stream>>>(h, W_pos, W_neg, x_pos, x_neg);

    // 3) degrees + dinv
    degree_k<<<2048, 256, 0, stream>>>(eip, dinv_pos);
    degree_k<<<2048, 256, 0, stream>>>(ein, dinv_neg);
    const int nb = (N_NODES + 255) / 256;
    rsqrt_k<<<nb, 256, 0, stream>>>(dinv_pos, N_NODES);
    rsqrt_k<<<nb, 256, 0, stream>>>(dinv_neg, N_NODES);

    // 4) edge scatter (wave per edge)
    scatter_k<<<4096, 256, 0, stream>>>(x_pos, dinv_pos, eip, agg_pos);
    scatter_k<<<4096, 256, 0, stream>>>(x_neg, dinv_neg, ein, agg_neg);

    // 5) fused: self-loop + GCN bias + psi GEMMs + psi bias -> delta_h
    gemm_final<<<782, 256, 0, stream>>>(agg_pos, agg_neg, x_pos, x_neg,
                                        dinv_pos, dinv_neg, b_pos, b_neg,
                                        W_psi_pos, W_psi_neg,
                                        b_psi_pos, b_psi_neg, out);
}